// net_28252294873195
// MI455X (gfx1250) — compile-verified
//
#include <hip/hip_runtime.h>
#include <hip/hip_bf16.h>
#include <math.h>

#define N_NODES 3072
#define E_EDGES 98304
#define CUM 36
#define HID 64
#define OUT_DIM 5
#define RCOLS (N_NODES * 9)   // 27648 floats per R row
#define RROWS 4               // R rows produced per block (amortize B reads)

typedef __attribute__((ext_vector_type(2))) float v2f;
typedef __attribute__((ext_vector_type(4))) float f32x4;
typedef __attribute__((ext_vector_type(8))) float v8f;

// ---------------------------------------------------------------- fast tanh
__device__ __forceinline__ float fast_tanh(float x) {
#if __has_builtin(__builtin_amdgcn_tanhf)
    return __builtin_amdgcn_tanhf(x);     // gfx1250 native v_tanh_f32 (TRANS)
#else
    float ax = fabsf(x);
    float e  = __expf(-2.0f * ax);
    float t  = (1.0f - e) / (1.0f + e);
    return copysignf(t, x);
#endif
}

__device__ __forceinline__ v8f wmma4(v2f a, v2f b, v8f c) {
    // V_WMMA_F32_16X16X4_F32 : exact f32 MAC, D = A(16x4) * B(4x16) + C
    return __builtin_amdgcn_wmma_f32_16x16x4_f32(
        /*neg_a=*/false, a, /*neg_b=*/false, b,
        /*c_mod=*/(short)0, c, /*reuse_a=*/false, /*reuse_b=*/false);
}

// ---------------------------------------------------------------- zero scratch
__global__ void zero_kernel(float* __restrict__ p, int n) {
    int i = blockIdx.x * blockDim.x + threadIdx.x;
    if (i < n) p[i] = 0.0f;
}

// ---------------------------------------------------------------- sheaf rank factors
// A'[n,j] = sum_d x[n,d]*Ws[d,j] + b_sheaf[j]   (bias folded into A)
// B [n,j] = sum_d x[n,d]*Ws[3+d,j]
__global__ void sheaf_ab_kernel(const float* __restrict__ x,
                                const float* __restrict__ Ws,
                                const float* __restrict__ bs,
                                float* __restrict__ A,
                                float* __restrict__ B) {
    int t = blockIdx.x * blockDim.x + threadIdx.x;
    if (t >= N_NODES * 9) return;
    int n = t / 9, j = t - n * 9;
    float x0 = x[n * 3 + 0], x1 = x[n * 3 + 1], x2 = x[n * 3 + 2];
    A[t] = fmaf(x0, Ws[0 * 9 + j], fmaf(x1, Ws[1 * 9 + j], fmaf(x2, Ws[2 * 9 + j], bs[j])));
    B[t] = fmaf(x0, Ws[3 * 9 + j], fmaf(x1, Ws[4 * 9 + j], x2 * Ws[5 * 9 + j]));
}

// ---------------------------------------------------------------- R tensor (HBM-write bound)
// For row-major R[i][j][c], the inner flat index t = j*9+c addresses B_flat[t]
// directly, so each row is tanh(ac[i][t%9] + B_flat[t]). B is L2-resident and
// read once per RROWS output rows; output streams via NT b128 stores.
__global__ __launch_bounds__(256) void r_kernel(const float* __restrict__ A,
                                                const float* __restrict__ B,
                                                float* __restrict__ R) {
    int i0 = blockIdx.x * RROWS;
    __shared__ float ac[RROWS][9];
    if (threadIdx.x < RROWS * 9)
        ac[threadIdx.x / 9][threadIdx.x % 9] = A[i0 * 9 + threadIdx.x];
    __syncthreads();
    const f32x4* Bv = (const f32x4*)B;
    float* dst0 = R + (size_t)i0 * RCOLS;
    for (int q = threadIdx.x; q < RCOLS / 4; q += 256) {
        f32x4 b = Bv[q];
        int base = q * 4;
        int c0 = base % 9;
#pragma unroll
        for (int rr = 0; rr < RROWS; ++rr) {
            int c = c0;
            f32x4 r;
            r.x = fast_tanh(ac[rr][c] + b.x); c = (c == 8) ? 0 : c + 1;
            r.y = fast_tanh(ac[rr][c] + b.y); c = (c == 8) ? 0 : c + 1;
            r.z = fast_tanh(ac[rr][c] + b.z); c = (c == 8) ? 0 : c + 1;
            r.w = fast_tanh(ac[rr][c] + b.w);
            __builtin_nontemporal_store(r, (f32x4*)(dst0 + (size_t)rr * RCOLS + base));
        }
    }
}

// ---------------------------------------------------------------- hop 1: x[3] -> h1[9], + degree
__global__ void hop1_kernel(const float* __restrict__ x, const int* __restrict__ ei,
                            const float* __restrict__ K, float* __restrict__ h1,
                            float* __restrict__ deg) {
    int e = blockIdx.x * blockDim.x + threadIdx.x;
    if (e >= E_EDGES) return;
    int s = ei[e], t = ei[E_EDGES + e];
    float x0 = x[s * 3 + 0], x1 = x[s * 3 + 1], x2 = x[s * 3 + 2];
    float* ht = h1 + t * 9;
#pragma unroll
    for (int k = 0; k < 3; ++k) {
        float kv = K[k * E_EDGES + e];
        atomicAdd(ht + k * 3 + 0, kv * x0);
        atomicAdd(ht + k * 3 + 1, kv * x1);
        atomicAdd(ht + k * 3 + 2, kv * x2);
    }
    atomicAdd(deg + t, 1.0f);
}

// ---------------------------------------------------------------- hop 2: h1[9] -> h2[27]
__global__ void hop2_kernel(const float* __restrict__ h1, const int* __restrict__ ei,
                            const float* __restrict__ K, float* __restrict__ h2) {
    int e = blockIdx.x * blockDim.x + threadIdx.x;
    if (e >= E_EDGES) return;
    int s = ei[e], t = ei[E_EDGES + e];
    float hv[9];
#pragma unroll
    for (int c = 0; c < 9; ++c) hv[c] = h1[s * 9 + c];
    float* ht = h2 + t * 27;
#pragma unroll
    for (int k = 0; k < 3; ++k) {
        float kv = K[k * E_EDGES + e];
#pragma unroll
        for (int c = 0; c < 9; ++c) atomicAdd(ht + k * 9 + c, kv * hv[c]);
    }
}

// ---------------------------------------------------------------- GCN aggregate of concat(h1,h2)
__global__ void agg_kernel(const float* __restrict__ h1, const float* __restrict__ h2,
                           const int* __restrict__ ei, float* __restrict__ agg) {
    int e = blockIdx.x * blockDim.x + threadIdx.x;
    if (e >= E_EDGES) return;
    int s = ei[e], t = ei[E_EDGES + e];
    float* at = agg + t * CUM;
#pragma unroll
    for (int c = 0; c < 9; ++c)  atomicAdd(at + c,     h1[s * 9 + c]);
#pragma unroll
    for (int c = 0; c < 27; ++c) atomicAdd(at + 9 + c, h2[s * 27 + c]);
}

// ---------------------------------------------------------------- fused dense head (WMMA f32)
// one wave32 per 16-row tile: relu((agg/deg)@Wc+bc) -> relu(@W1+b1) -> @W2+b2
// Weights that need column padding are staged zero-padded in LDS so every
// B-fragment read is an unconditional ds_load (no EXEC churn between WMMAs).
__global__ __launch_bounds__(32) void dense_head_kernel(
    const float* __restrict__ agg, const float* __restrict__ deg,
    const float* __restrict__ Wc,  const float* __restrict__ bc,
    const float* __restrict__ W1,  const float* __restrict__ b1,
    const float* __restrict__ W2,  const float* __restrict__ b2,
    float* __restrict__ out) {
    int row0 = blockIdx.x * 16;
    int lane = threadIdx.x;
    int hi = lane >> 4, lo = lane & 15;

    __shared__ float rdeg[16];
    __shared__ float Wcp[CUM][48];   // Wc zero-padded to 48 cols
    __shared__ float W2p[HID][16];   // W2 zero-padded to 16 cols
    __shared__ float t1[16][48];     // conv tile (36 valid cols; pad cols never read)
    __shared__ float t2[16][64];     // hidden tile

    for (int idx = lane; idx < CUM * 48; idx += 32) {
        int rr = idx / 48, cc = idx - rr * 48;
        Wcp[rr][cc] = (cc < CUM) ? Wc[rr * CUM + cc] : 0.0f;
    }
    for (int idx = lane; idx < HID * 16; idx += 32) {
        int rr = idx >> 4, cc = idx & 15;
        W2p[rr][cc] = (cc < OUT_DIM) ? W2[rr * OUT_DIM + cc] : 0.0f;
    }
    if (lane < 16) rdeg[lane] = 1.0f / fmaxf(deg[row0 + lane], 1.0f);
    __syncthreads();

    // ---- stage 1: [16x36] x [36x48(pad)] -> 3 col-tiles of 16
    v8f c0 = {}, c1 = {}, c2 = {};
    const float* arow = agg + (size_t)(row0 + lo) * CUM;
    float rd = rdeg[lo];
#pragma unroll
    for (int kk = 0; kk < 9; ++kk) {
        int k0 = kk * 4 + 2 * hi;                       // A frag: lane holds K=k0,k0+1
        v2f a; a.x = arow[k0] * rd; a.y = arow[k0 + 1] * rd;
        v2f b;
        b.x = Wcp[k0][lo];       b.y = Wcp[k0 + 1][lo];       c0 = wmma4(a, b, c0);
        b.x = Wcp[k0][16 + lo];  b.y = Wcp[k0 + 1][16 + lo];  c1 = wmma4(a, b, c1);
        b.x = Wcp[k0][32 + lo];  b.y = Wcp[k0 + 1][32 + lo];  c2 = wmma4(a, b, c2);
    }
#pragma unroll
    for (int r = 0; r < 8; ++r) {                       // D frag: row = r + 8*hi, col = lo
        int m = r + 8 * hi;
        int n2 = 32 + lo;
        float bias2 = (n2 < CUM) ? bc[n2] : 0.0f;
        t1[m][lo]      = fmaxf(c0[r] + bc[lo], 0.0f);
        t1[m][16 + lo] = fmaxf(c1[r] + bc[16 + lo], 0.0f);
        t1[m][n2]      = fmaxf(c2[r] + bias2, 0.0f);    // pad cols written, never read
    }
    __syncthreads();

    // ---- stage 2: [16x36] x [36x64] -> 4 col-tiles
    v8f d0 = {}, d1 = {}, d2 = {}, d3 = {};
#pragma unroll
    for (int kk = 0; kk < 9; ++kk) {
        int k0 = kk * 4 + 2 * hi;
        v2f a; a.x = t1[lo][k0]; a.y = t1[lo][k0 + 1];
        v2f b;
        b.x = W1[k0 * HID + lo];       b.y = W1[(k0 + 1) * HID + lo];       d0 = wmma4(a, b, d0);
        b.x = W1[k0 * HID + 16 + lo];  b.y = W1[(k0 + 1) * HID + 16 + lo];  d1 = wmma4(a, b, d1);
        b.x = W1[k0 * HID + 32 + lo];  b.y = W1[(k0 + 1) * HID + 32 + lo];  d2 = wmma4(a, b, d2);
        b.x = W1[k0 * HID + 48 + lo];  b.y = W1[(k0 + 1) * HID + 48 + lo];  d3 = wmma4(a, b, d3);
    }
#pragma unroll
    for (int r = 0; r < 8; ++r) {
        int m = r + 8 * hi;
        t2[m][lo]      = fmaxf(d0[r] + b1[lo], 0.0f);
        t2[m][16 + lo] = fmaxf(d1[r] + b1[16 + lo], 0.0f);
        t2[m][32 + lo] = fmaxf(d2[r] + b1[32 + lo], 0.0f);
        t2[m][48 + lo] = fmaxf(d3[r] + b1[48 + lo], 0.0f);
    }
    __syncthreads();

    // ---- stage 3: [16x64] x [64x16(pad)] (cols 5..15 zero)
    v8f o = {};
#pragma unroll
    for (int kk = 0; kk < 16; ++kk) {
        int k0 = kk * 4 + 2 * hi;
        v2f a; a.x = t2[lo][k0]; a.y = t2[lo][k0 + 1];
        v2f b; b.x = W2p[k0][lo]; b.y = W2p[k0 + 1][lo];
        o = wmma4(a, b, o);
    }
    if (lo < OUT_DIM) {
#pragma unroll
        for (int r = 0; r < 8; ++r) {
            int m = r + 8 * hi;
            out[(size_t)(row0 + m) * OUT_DIM + lo] = o[r] + b2[lo];
        }
    }
}

// ---------------------------------------------------------------- launcher
extern "C" void kernel_launch(void* const* d_in, const int* in_sizes, int n_in,
                              void* d_out, int out_size, void* d_ws, size_t ws_size,
                              hipStream_t stream) {
    const float* x   = (const float*)d_in[0];
    const int*   ei  = (const int*)  d_in[1];
    const float* K   = (const float*)d_in[2];
    const float* Ws  = (const float*)d_in[3];
    const float* bs  = (const float*)d_in[4];
    const float* Wc  = (const float*)d_in[5];
    const float* bc  = (const float*)d_in[6];
    const float* W1  = (const float*)d_in[7];
    const float* b1  = (const float*)d_in[8];
    const float* W2  = (const float*)d_in[9];
    const float* b2  = (const float*)d_in[10];

    float* out = (float*)d_out;                       // [N,5]
    float* R   = out + N_NODES * OUT_DIM;             // [N,N,9] (offset 16B-aligned)

    float* ws  = (float*)d_ws;                        // all offsets 16B-aligned
    float* A   = ws;                                  // 27648
    float* B   = ws + 27648;                          // 27648
    float* h1  = ws + 55296;                          // 27648  (zeroed)
    float* h2  = ws + 82944;                          // 82944  (zeroed)
    float* agg = ws + 165888;                         // 110592 (zeroed)
    float* deg = ws + 276480;                         // 3072   (zeroed)
    const int nzero = 224256;                         // h1..deg contiguous

    zero_kernel<<<(nzero + 255) / 256, 256, 0, stream>>>(h1, nzero);
    sheaf_ab_kernel<<<(N_NODES * 9 + 255) / 256, 256, 0, stream>>>(x, Ws, bs, A, B);
    r_kernel<<<N_NODES / RROWS, 256, 0, stream>>>(A, B, R);
    hop1_kernel<<<(E_EDGES + 255) / 256, 256, 0, stream>>>(x, ei, K, h1, deg);
    hop2_kernel<<<(E_EDGES + 255) / 256, 256, 0, stream>>>(h1, ei, K, h2);
    agg_kernel<<<(E_EDGES + 255) / 256, 256, 0, stream>>>(h1, h2, ei, agg);
    dense_head_kernel<<<N_NODES / 16, 32, 0, stream>>>(agg, deg, Wc, bc, W1, b1, W2, b2, out);
}